// FFPS_Sampler_54898271977792
// MI455X (gfx1250) — compile-verified
//
#include <hip/hip_runtime.h>

typedef __attribute__((ext_vector_type(2))) float v2f;
typedef __attribute__((ext_vector_type(8))) float v8f;

#define NPTS     8192
#define CH       68          // 3 coords + 64 feature channels + 1 zero pad (K multiple of 4)
#define NTHREADS 1024
#define PER_THREAD (NPTS / NTHREADS)

// ---------------------------------------------------------------------------
// Kernel 1: pack [points | features^T] into B x N x 68 row-major + squared norms
// ---------------------------------------------------------------------------
__global__ void pack_norms_kernel(const float* __restrict__ points,
                                  const float* __restrict__ features,
                                  float* __restrict__ Fpk,
                                  float* __restrict__ sq,
                                  int N, int C) {
  int idx = blockIdx.x * blockDim.x + threadIdx.x;   // b*N + n
  int b = idx / N;
  int n = idx - b * N;
  float s = 0.f;
  float* dst = Fpk + (size_t)idx * CH;
  const float* p = points + (size_t)idx * 3;
#pragma unroll
  for (int c = 0; c < 3; ++c) { float v = p[c]; dst[c] = v; s += v * v; }
  const float* f = features + (size_t)b * C * N + n;
  for (int c = 0; c < C; ++c) { float v = f[(size_t)c * N]; dst[3 + c] = v; s += v * v; }
  dst[CH - 1] = 0.f;   // K pad so 68 % 4 == 0
  sq[idx] = s;
}

// ---------------------------------------------------------------------------
// Kernel 2: dist[b,i,j] = sq[i] + sq[j] - 2 * <F_i, F_j>  via V_WMMA_F32_16X16X4_F32
// One wave per 16x16 output tile, 17 WMMA (K=68) per tile. EXEC all-1s by construction.
// A 16x4 f32 fragment layout (ISA 7.12.2): lanes 0-15 -> M=lane, VGPR0/1 = K+0,K+1;
// lanes 16-31 -> same M, VGPR0/1 = K+2,K+3  => one aligned float2 load per step.
// B fragment mirrors with N in place of M (Gram matrix: same loader, j-tile base).
// ---------------------------------------------------------------------------
__global__ void dist_wmma_kernel(const float* __restrict__ Fpk,
                                 const float* __restrict__ sq,
                                 float* __restrict__ dist, int N) {
  const int lane = threadIdx.x & 31;
  const int wave = threadIdx.x >> 5;           // 0..7
  const int b    = blockIdx.z;
  const int i0   = blockIdx.y * 16;
  const int j0   = (blockIdx.x * 8 + wave) * 16;
  const int half = lane >> 4;                  // 0: K,K+1   1: K+2,K+3
  const int l15  = lane & 15;

  const float* F    = Fpk + (size_t)b * N * CH;
  const float* arow = F + (size_t)(i0 + l15) * CH + 2 * half;
  const float* brow = F + (size_t)(j0 + l15) * CH + 2 * half;

  v8f acc = {};
#pragma unroll
  for (int k = 0; k < CH; k += 4) {
    v2f a  = *(const v2f*)(arow + k);
    v2f bb = *(const v2f*)(brow + k);
    acc = __builtin_amdgcn_wmma_f32_16x16x4_f32(
        /*neg_a=*/false, a, /*neg_b=*/false, bb,
        /*c_mod=*/(short)0, acc, /*reuse_a=*/false, /*reuse_b=*/false);
  }

  // C/D layout: lane (0-15|16-31) holds N=l15, VGPR v holds M = v (+8 for high half)
  const float* sqb = sq + (size_t)b * N;
  float sqj = sqb[j0 + l15];
#pragma unroll
  for (int v = 0; v < 8; ++v) {
    int row = i0 + half * 8 + v;
    float d = sqb[row] + sqj - 2.0f * acc[v];
    dist[((size_t)b * N + row) * N + (j0 + l15)] = d;
  }
}

// ---------------------------------------------------------------------------
// Kernel 3: sequential FPS loop. One workgroup (32 wave32s) per batch.
// mind[] lives in LDS (32KB of 320KB). Argmax = per-thread scan (8 strided elems)
// -> wave shuffle reduce -> 32 partials -> wave-0 shuffle reduce. First-occurrence
// tie-break (val equal -> smaller index) matches jnp.argmax.
// mode 0: read precomputed dist rows (HBM). mode 1: recompute row from L2-resident Fpk.
// ---------------------------------------------------------------------------
__global__ __launch_bounds__(NTHREADS)
void fps_kernel(const float* __restrict__ Fpk,
                const float* __restrict__ sq,
                const float* __restrict__ dist,
                int* __restrict__ out,
                int N, int npoint, int mode) {
  __shared__ float mind_s[NPTS];
  __shared__ float rv[32];
  __shared__ int   ri[32];
  __shared__ float ffar[CH];
  __shared__ float s_sqfar;
  __shared__ int   s_far;

  const int b    = blockIdx.x;
  const int t    = threadIdx.x;
  const int lane = t & 31;
  const int wv   = t >> 5;
  const float* sqb = sq + (size_t)b * N;
  const float* Fb  = Fpk + (size_t)b * N * CH;

  int far = 0;
  if (t == 0) out[(size_t)b * npoint] = 0;

  for (int k = 0; k < npoint; ++k) {
    if (k > 0) {
      // ---- argmax over mind_s ----
      float bv = -3.4e38f; int bi = N;
#pragma unroll
      for (int u = 0; u < PER_THREAD; ++u) {
        int j = t + u * NTHREADS;
        float v = mind_s[j];
        if (v > bv) { bv = v; bi = j; }
      }
#pragma unroll
      for (int off = 16; off > 0; off >>= 1) {
        float ov = __shfl_down(bv, off);
        int   oi = __shfl_down(bi, off);
        if (ov > bv || (ov == bv && oi < bi)) { bv = ov; bi = oi; }
      }
      if (lane == 0) { rv[wv] = bv; ri[wv] = bi; }
      __syncthreads();
      if (wv == 0) {
        bv = rv[lane]; bi = ri[lane];
#pragma unroll
        for (int off = 16; off > 0; off >>= 1) {
          float ov = __shfl_down(bv, off);
          int   oi = __shfl_down(bi, off);
          if (ov > bv || (ov == bv && oi < bi)) { bv = ov; bi = oi; }
        }
        if (lane == 0) { s_far = bi; out[(size_t)b * npoint + k] = bi; }
      }
      __syncthreads();
      far = s_far;
    }

    // ---- fold row(far) into mind ----
    if (mode == 0) {
      const float* row = dist + ((size_t)b * N + far) * N;
      if (k == 0) {
#pragma unroll
        for (int u = 0; u < PER_THREAD; ++u) { int j = t + u * NTHREADS; mind_s[j] = row[j]; }
      } else {
#pragma unroll
        for (int u = 0; u < PER_THREAD; ++u) { int j = t + u * NTHREADS; mind_s[j] = fminf(mind_s[j], row[j]); }
      }
    } else {
      if (t < CH)  ffar[t]  = Fb[(size_t)far * CH + t];
      if (t == 0)  s_sqfar  = sqb[far];
      __syncthreads();
      float sqf = s_sqfar;
#pragma unroll
      for (int u = 0; u < PER_THREAD; ++u) {
        int j = t + u * NTHREADS;
        const float* fj = Fb + (size_t)j * CH;
        float dot = 0.f;
#pragma unroll 4
        for (int c = 0; c < CH; ++c) dot += ffar[c] * fj[c];
        float d = sqf + sqb[j] - 2.f * dot;
        mind_s[j] = (k == 0) ? d : fminf(mind_s[j], d);
      }
    }
    __syncthreads();
  }
}

// ---------------------------------------------------------------------------
// Host launcher
// ---------------------------------------------------------------------------
extern "C" void kernel_launch(void* const* d_in, const int* in_sizes, int n_in,
                              void* d_out, int out_size, void* d_ws, size_t ws_size,
                              hipStream_t stream) {
  const int B = 2, N = NPTS, C = 64;
  const int npoint = out_size / B;                 // 1024

  const float* points   = (const float*)d_in[0];
  const float* features = (const float*)d_in[1];
  int* out = (int*)d_out;

  char* ws = (char*)d_ws;
  float* Fpk = (float*)ws;
  const size_t szF  = (size_t)B * N * CH * sizeof(float);
  float* sq  = (float*)(ws + szF);
  const size_t szSq = (size_t)B * N * sizeof(float);
  const size_t offDist = (szF + szSq + 255) & ~(size_t)255;
  const size_t need    = offDist + (size_t)B * N * N * sizeof(float);
  float* dist = (float*)(ws + offDist);

  // Deterministic: depends only on ws_size (constant across calls).
  const int mode = (ws_size >= need) ? 0 : 1;

  pack_norms_kernel<<<(B * N) / 256, 256, 0, stream>>>(points, features, Fpk, sq, N, C);

  if (mode == 0) {
    dim3 grid(N / 16 / 8, N / 16, B);              // 64 x 512 x 2, 8 tiles per block
    dist_wmma_kernel<<<grid, 256, 0, stream>>>(Fpk, sq, dist, N);
  }

  fps_kernel<<<B, NTHREADS, 0, stream>>>(Fpk, sq, dist, out, N, npoint, mode);
}